// GAT_16733192585727
// MI455X (gfx1250) — compile-verified
//
#include <hip/hip_runtime.h>
#include <math.h>

// ---------------- problem constants (fixed; not checked against in_sizes) ---
#define IN_DIM   128
#define HID      128          // HEADS * OUT_C
#define HEADS    8
#define OUT_C    16
#define NEG_SLOPE 0.2f

typedef __attribute__((ext_vector_type(2))) float v2f;
typedef __attribute__((ext_vector_type(8))) float v8f;

// ---- order-preserving float<->uint encoding (for atomicMax on u32) ----------
__device__ __forceinline__ unsigned ford(float f) {
    unsigned u = __float_as_uint(f);
    return (u & 0x80000000u) ? ~u : (u | 0x80000000u);
}
__device__ __forceinline__ float unford(unsigned o) {
    return (o & 0x80000000u) ? __uint_as_float(o & 0x7fffffffu)
                             : __uint_as_float(~o);
}
__device__ __forceinline__ float lrelu(float e) {
    return e > 0.0f ? e : NEG_SLOPE * e;
}
// edge i in [0, E+N): first E from edge_index, rest are self loops
__device__ __forceinline__ void edge_sd(const int* __restrict__ ei, int e,
                                        int idx, int& s, int& d) {
    if (idx < e) { s = ei[idx]; d = ei[e + idx]; }
    else         { s = idx - e; d = idx - e; }
}

// ---------------- 0) zero/neg-inf init of all accumulators -------------------
__global__ __launch_bounds__(256) void init_ws(float* acc1, unsigned* m1, float* s1,
                                               float* acc2, unsigned* m2, float* s2,
                                               int n) {
    long long i = (long long)blockIdx.x * blockDim.x + threadIdx.x;
    if (i < (long long)n * HID) acc1[i] = 0.0f;
    if (i < (long long)n * HEADS) { m1[i] = 0u; s1[i] = 0.0f; }  // 0 < ford(-inf)
    if (i < (long long)n * 2) acc2[i] = 0.0f;
    if (i < n) { m2[i] = 0u; s2[i] = 0.0f; }
}

// ---------------- 1) h1 = X @ W1 via native FP32 WMMA 16x16x4 ---------------
// One wave per 16x16 output tile. Tiles: (n/16) x (HID/16=8). 8 waves/block.
__global__ __launch_bounds__(256) void gemm1_wmma(const float* __restrict__ X,
                                                  const float* __restrict__ W,
                                                  float* __restrict__ H, int n) {
    const int lane   = threadIdx.x & 31;
    const int waveId = blockIdx.x * 8 + (threadIdx.x >> 5);
    const int mtiles = n >> 4;
    const int tile_m = waveId >> 3;          // over rows of X
    const int tile_n = waveId & 7;           // 8 column tiles of W1
    if (tile_m >= mtiles) return;

    const int r16    = lane & 15;            // A row / B,C column within tile
    const int hiHalf = lane >> 4;            // lanes 16..31 take K+2 / M+8
    const int m0 = tile_m * 16;
    const int n0 = tile_n * 16;

    const float* __restrict__ xrow = X + (long long)(m0 + r16) * IN_DIM;
    v8f c = {};
    #pragma unroll 4
    for (int k0 = 0; k0 < IN_DIM; k0 += 4) {
        const int ka = k0 + hiHalf * 2;
        v2f a, b;
        a.x = xrow[ka];
        a.y = xrow[ka + 1];
        b.x = W[(long long)ka * HID + n0 + r16];
        b.y = W[(long long)(ka + 1) * HID + n0 + r16];
        // D = A(16x4) * B(4x16) + C   (fp32 in, fp32 acc)
        c = __builtin_amdgcn_wmma_f32_16x16x4_f32(false, a, false, b,
                                                  (short)0, c, false, false);
    }
    // C/D layout: VGPR r -> M = r + 8*hiHalf, N = lane&15
    float* __restrict__ hout = H + (long long)(m0 + hiHalf * 8) * HID + n0 + r16;
    #pragma unroll
    for (int r = 0; r < 8; ++r)
        hout[(long long)r * HID] = c[r];
}

// ---------------- 2) per-node attention scalars a_src/a_dst ------------------
__global__ __launch_bounds__(256) void attn_coef1(const float* __restrict__ H,
                                                  const float* __restrict__ att_src,
                                                  const float* __restrict__ att_dst,
                                                  float* __restrict__ asrc,
                                                  float* __restrict__ adst, int n) {
    int idx = blockIdx.x * blockDim.x + threadIdx.x;    // over n*HEADS
    if (idx >= n * HEADS) return;
    const int node = idx / HEADS, h = idx % HEADS;
    const float* __restrict__ hv = H + (long long)node * HID + h * OUT_C;
    const float* __restrict__ as = att_src + h * OUT_C;
    const float* __restrict__ ad = att_dst + h * OUT_C;
    float s = 0.0f, d = 0.0f;
    #pragma unroll
    for (int i = 0; i < OUT_C; ++i) { s += hv[i] * as[i]; d += hv[i] * ad[i]; }
    asrc[idx] = s; adst[idx] = d;
}

// ---------------- 3) segment max over incoming edges (per head) --------------
__global__ __launch_bounds__(256) void edge_max1(const int* __restrict__ ei, int e, int et,
                                                 const float* __restrict__ asrc,
                                                 const float* __restrict__ adst,
                                                 unsigned* __restrict__ m) {
    long long idx = (long long)blockIdx.x * blockDim.x + threadIdx.x;
    if (idx >= (long long)et * HEADS) return;
    int eidx = (int)(idx >> 3), h = (int)(idx & 7);
    int src, dst; edge_sd(ei, e, eidx, src, dst);
    float ev = lrelu(asrc[src * HEADS + h] + adst[dst * HEADS + h]);
    atomicMax(&m[dst * HEADS + h], ford(ev));
}

// ---------------- 4) segment sum of exp(e - m) -------------------------------
__global__ __launch_bounds__(256) void edge_sum1(const int* __restrict__ ei, int e, int et,
                                                 const float* __restrict__ asrc,
                                                 const float* __restrict__ adst,
                                                 const unsigned* __restrict__ m,
                                                 float* __restrict__ s) {
    long long idx = (long long)blockIdx.x * blockDim.x + threadIdx.x;
    if (idx >= (long long)et * HEADS) return;
    int eidx = (int)(idx >> 3), h = (int)(idx & 7);
    int src, dst; edge_sd(ei, e, eidx, src, dst);
    float ev = lrelu(asrc[src * HEADS + h] + adst[dst * HEADS + h]);
    atomicAdd(&s[dst * HEADS + h], __expf(ev - unford(m[dst * HEADS + h])));
}

// ---------------- 5) message pass: one wave per edge, float4 per lane --------
__global__ __launch_bounds__(256) void edge_msg1(const int* __restrict__ ei, int e, int et,
                                                 const float* __restrict__ asrc,
                                                 const float* __restrict__ adst,
                                                 const unsigned* __restrict__ m,
                                                 const float* __restrict__ s,
                                                 const float* __restrict__ h1,
                                                 float* __restrict__ acc) {
    const int lane = threadIdx.x & 31;
    long long eidx = (long long)blockIdx.x * 8 + (threadIdx.x >> 5);
    if (eidx >= et) return;
    int src, dst; edge_sd(ei, e, (int)eidx, src, dst);
    const int h = lane >> 2;                     // 4 lanes per head (16 ch each)
    float ev = lrelu(asrc[src * HEADS + h] + adst[dst * HEADS + h]);
    float alpha = __expf(ev - unford(m[dst * HEADS + h])) / s[dst * HEADS + h];
    const float4 v = ((const float4*)(h1 + (long long)src * HID))[lane];
    float* __restrict__ out = acc + (long long)dst * HID + lane * 4;
    atomicAdd(out + 0, alpha * v.x);
    atomicAdd(out + 1, alpha * v.y);
    atomicAdd(out + 2, alpha * v.z);
    atomicAdd(out + 3, alpha * v.w);
}

// ---------------- 6) bias + ELU, write layer-2 input (reuses h1 buffer) ------
__global__ __launch_bounds__(256) void finalize1(const float* __restrict__ acc,
                                                 const float* __restrict__ b1,
                                                 float* __restrict__ h2, int n) {
    long long i = (long long)blockIdx.x * blockDim.x + threadIdx.x;
    if (i >= (long long)n * HID) return;
    float v = acc[i] + b1[(int)(i & (HID - 1))];
    h2[i] = v > 0.0f ? v : (__expf(v) - 1.0f);   // ELU
}

// ---------------- 7) layer-2 projection (128->2) + attention scalars ---------
__global__ __launch_bounds__(256) void layer2_proj(const float* __restrict__ h2,
                                                   const float* __restrict__ W2,
                                                   const float* __restrict__ as2,
                                                   const float* __restrict__ ad2,
                                                   float* __restrict__ g,
                                                   float* __restrict__ asrc,
                                                   float* __restrict__ adst, int n) {
    int i = blockIdx.x * blockDim.x + threadIdx.x;
    if (i >= n) return;
    const float4* __restrict__ hv = (const float4*)(h2 + (long long)i * HID);
    float g0 = 0.0f, g1 = 0.0f;
    #pragma unroll 8
    for (int k = 0; k < HID / 4; ++k) {
        float4 v = hv[k];
        const float* w = W2 + k * 8;             // W2 row-major [128][2]
        g0 += v.x * w[0] + v.y * w[2] + v.z * w[4] + v.w * w[6];
        g1 += v.x * w[1] + v.y * w[3] + v.z * w[5] + v.w * w[7];
    }
    g[i * 2] = g0; g[i * 2 + 1] = g1;
    asrc[i] = g0 * as2[0] + g1 * as2[1];
    adst[i] = g0 * ad2[0] + g1 * ad2[1];
}

// ---------------- 8-10) layer-2 softmax passes (H=1) -------------------------
__global__ __launch_bounds__(256) void edge_max2(const int* __restrict__ ei, int e, int et,
                                                 const float* __restrict__ asrc,
                                                 const float* __restrict__ adst,
                                                 unsigned* __restrict__ m) {
    int idx = blockIdx.x * blockDim.x + threadIdx.x;
    if (idx >= et) return;
    int src, dst; edge_sd(ei, e, idx, src, dst);
    atomicMax(&m[dst], ford(lrelu(asrc[src] + adst[dst])));
}
__global__ __launch_bounds__(256) void edge_sum2(const int* __restrict__ ei, int e, int et,
                                                 const float* __restrict__ asrc,
                                                 const float* __restrict__ adst,
                                                 const unsigned* __restrict__ m,
                                                 float* __restrict__ s) {
    int idx = blockIdx.x * blockDim.x + threadIdx.x;
    if (idx >= et) return;
    int src, dst; edge_sd(ei, e, idx, src, dst);
    float ev = lrelu(asrc[src] + adst[dst]);
    atomicAdd(&s[dst], __expf(ev - unford(m[dst])));
}
__global__ __launch_bounds__(256) void edge_msg2(const int* __restrict__ ei, int e, int et,
                                                 const float* __restrict__ asrc,
                                                 const float* __restrict__ adst,
                                                 const unsigned* __restrict__ m,
                                                 const float* __restrict__ s,
                                                 const float* __restrict__ g,
                                                 float* __restrict__ acc) {
    int idx = blockIdx.x * blockDim.x + threadIdx.x;
    if (idx >= et) return;
    int src, dst; edge_sd(ei, e, idx, src, dst);
    float ev = lrelu(asrc[src] + adst[dst]);
    float alpha = __expf(ev - unford(m[dst])) / s[dst];
    atomicAdd(&acc[dst * 2 + 0], alpha * g[src * 2 + 0]);
    atomicAdd(&acc[dst * 2 + 1], alpha * g[src * 2 + 1]);
}

// ---------------- 11) bias + log_softmax over the 2 classes ------------------
__global__ __launch_bounds__(256) void finalize_out(const float* __restrict__ acc2,
                                                    const float* __restrict__ b2,
                                                    float* __restrict__ out, int n) {
    int i = blockIdx.x * blockDim.x + threadIdx.x;
    if (i >= n) return;
    float v0 = acc2[i * 2] + b2[0];
    float v1 = acc2[i * 2 + 1] + b2[1];
    float mx = fmaxf(v0, v1);
    float l  = __logf(__expf(v0 - mx) + __expf(v1 - mx));
    out[i * 2]     = v0 - mx - l;
    out[i * 2 + 1] = v1 - mx - l;
}

// ======================== host-side launcher =================================
extern "C" void kernel_launch(void* const* d_in, const int* in_sizes, int n_in,
                              void* d_out, int out_size, void* d_ws, size_t ws_size,
                              hipStream_t stream) {
    const float* x        = (const float*)d_in[0];
    const int*   ei       = (const int*)  d_in[1];
    const float* W1       = (const float*)d_in[2];
    const float* att_src1 = (const float*)d_in[3];
    const float* att_dst1 = (const float*)d_in[4];
    const float* b1       = (const float*)d_in[5];
    const float* W2       = (const float*)d_in[6];
    const float* att_src2 = (const float*)d_in[7];
    const float* att_dst2 = (const float*)d_in[8];
    const float* b2       = (const float*)d_in[9];
    float* out = (float*)d_out;

    const int n  = in_sizes[0] / IN_DIM;   // 50000
    const int e  = in_sizes[1] / 2;        // 1.6M
    const int et = e + n;                  // + self loops

    // ---- carve workspace (256B aligned slabs) ----
    char* p = (char*)d_ws;
    auto carve = [&](size_t bytes) { void* r = (void*)p;
                                     p += (bytes + 255) & ~(size_t)255; return r; };
    float*    h1    = (float*)   carve((size_t)n * HID * 4);  // also reused as h2
    float*    asrc1 = (float*)   carve((size_t)n * HEADS * 4);
    float*    adst1 = (float*)   carve((size_t)n * HEADS * 4);
    unsigned* m1    = (unsigned*)carve((size_t)n * HEADS * 4);
    float*    s1    = (float*)   carve((size_t)n * HEADS * 4);
    float*    acc1  = (float*)   carve((size_t)n * HID * 4);
    float*    g2    = (float*)   carve((size_t)n * 2 * 4);
    float*    asrc2 = (float*)   carve((size_t)n * 4);
    float*    adst2 = (float*)   carve((size_t)n * 4);
    unsigned* m2    = (unsigned*)carve((size_t)n * 4);
    float*    s2    = (float*)   carve((size_t)n * 4);
    float*    acc2  = (float*)   carve((size_t)n * 2 * 4);

    const int B = 256;
    auto cdiv = [](long long a, long long b) { return (unsigned)((a + b - 1) / b); };

    // 0) init accumulators (every call, for graph-replay determinism of state)
    init_ws<<<cdiv((long long)n * HID, B), B, 0, stream>>>(acc1, m1, s1, acc2, m2, s2, n);

    // 1) h1 = x @ W1  (fp32 WMMA; (n/16)*8 tiles, 8 waves/block)
    gemm1_wmma<<<cdiv((long long)(n / 16) * 8, 8), B, 0, stream>>>(x, W1, h1, n);

    // 2) attention scalars
    attn_coef1<<<cdiv((long long)n * HEADS, B), B, 0, stream>>>(h1, att_src1, att_dst1,
                                                                asrc1, adst1, n);
    // 3-5) segment softmax + aggregation, layer 1
    edge_max1<<<cdiv((long long)et * HEADS, B), B, 0, stream>>>(ei, e, et, asrc1, adst1, m1);
    edge_sum1<<<cdiv((long long)et * HEADS, B), B, 0, stream>>>(ei, e, et, asrc1, adst1, m1, s1);
    edge_msg1<<<cdiv((long long)et * 32, B), B, 0, stream>>>(ei, e, et, asrc1, adst1,
                                                             m1, s1, h1, acc1);
    // 6) bias + ELU -> h2 (reuse h1 slab)
    finalize1<<<cdiv((long long)n * HID, B), B, 0, stream>>>(acc1, b1, h1, n);

    // 7) layer-2 projection + attention scalars
    layer2_proj<<<cdiv(n, B), B, 0, stream>>>(h1, W2, att_src2, att_dst2, g2, asrc2, adst2, n);

    // 8-10) segment softmax + aggregation, layer 2
    edge_max2<<<cdiv(et, B), B, 0, stream>>>(ei, e, et, asrc2, adst2, m2);
    edge_sum2<<<cdiv(et, B), B, 0, stream>>>(ei, e, et, asrc2, adst2, m2, s2);
    edge_msg2<<<cdiv(et, B), B, 0, stream>>>(ei, e, et, asrc2, adst2, m2, s2, g2, acc2);

    // 11) bias + log_softmax
    finalize_out<<<cdiv(n, B), B, 0, stream>>>(acc2, b2, out, n);
}